// MultiplicityMasking_89421219102863
// MI455X (gfx1250) — compile-verified
//
#include <hip/hip_runtime.h>
#include <hip/hip_bf16.h>

// MultiplicityMasking on MI455X (gfx1250)
// Memory-bound: ~1.14 GB total traffic -> ~49us roofline at 23.3 TB/s.
// Quantile of integer-valued mult (0..28) done exactly via 29-bin histogram.
// CDNA5 paths used: global_load_async_to_lds_b128 (+ s_wait_asynccnt) for tile
// staging, nontemporal stores for the write-once output.

#define ROWS  128   // rows per block == block size
#define FEATS 85
#define NPART 28

// Low 32 bits of a generic pointer to an LDS object are the LDS byte offset
// (flat LDS aperture: addr = {SHARED_BASE, offset}).
__device__ __forceinline__ unsigned lds_off(const void* p) {
  return (unsigned)(unsigned long long)p;
}

__device__ __forceinline__ void async_copy_b128(unsigned lds, unsigned long long gaddr) {
  asm volatile("global_load_async_to_lds_b128 %0, %1, off"
               :: "v"(lds), "v"(gaddr) : "memory");
}

// ---------------- kernel 1: zero histogram ----------------
__global__ void mm_init(unsigned* __restrict__ hist) {
  if (threadIdx.x <= NPART) hist[threadIdx.x] = 0u;
}

// ---------------- kernel 2: per-row multiplicity histogram ----------------
__global__ void mm_hist(const float* __restrict__ x, unsigned* __restrict__ hist, int B) {
  __shared__ float    xt[ROWS * FEATS];      // 43520 B
  __shared__ unsigned lhist[NPART + 1];
  const int tid = threadIdx.x;
  const long long rowBase = (long long)blockIdx.x * ROWS;
  long long rem = (long long)B - rowBase;
  const int rows = rem < ROWS ? (int)rem : ROWS;

  if (tid <= NPART) lhist[tid] = 0u;

  if (rows == ROWS) {
    const unsigned long long src =
        (unsigned long long)(const void*)x + (unsigned long long)rowBase * (FEATS * 4ull);
    const unsigned dst = lds_off(xt);
    const int chunks = ROWS * FEATS * 4 / 16;   // 2720 b128 chunks
    for (int c = tid; c < chunks; c += ROWS)
      async_copy_b128(dst + (unsigned)c * 16u, src + (unsigned long long)c * 16u);
    __builtin_amdgcn_s_wait_asynccnt(0);
  } else {
    for (int i = tid; i < rows * FEATS; i += ROWS)
      xt[i] = x[rowBase * FEATS + i];
  }
  __syncthreads();

  if (tid < rows) {
    int cnt = 0;
#pragma unroll
    for (int k = 0; k < NPART; ++k)
      cnt += (xt[tid * FEATS + 1 + 3 * k] > 0.01f) ? 1 : 0;
    atomicAdd(&lhist[cnt], 1u);
  }
  __syncthreads();

  if (tid <= NPART) {
    unsigned c = lhist[tid];
    if (c) atomicAdd(&hist[tid], c);
  }
}

// ---------------- kernel 3: exact 75th percentile from histogram ----------------
__global__ void mm_thresh(const unsigned* __restrict__ hist, float* __restrict__ thr, int B) {
  if (threadIdx.x == 0 && blockIdx.x == 0) {
    double pos = 0.75 * (double)(B - 1);
    long long k = (long long)pos;
    double frac = pos - (double)k;
    long long cum = 0;
    int vlo = NPART, vhi = NPART;
    bool glo = false, ghi = false;
    for (int v = 0; v <= NPART; ++v) {
      cum += (long long)hist[v];
      if (!glo && cum > k)     { vlo = v; glo = true; }
      if (!ghi && cum > k + 1) { vhi = v; ghi = true; break; }
    }
    *thr = (float)((double)vlo + frac * ((double)vhi - (double)vlo));
  }
}

// ---------------- kernel 4: build masks and write output ----------------
__global__ void mm_apply(const float* __restrict__ x, const float* __restrict__ ru,
                         const float* __restrict__ thr_p, float* __restrict__ out, int B) {
  __shared__ float    xt[ROWS * FEATS];   // 43520 B
  __shared__ float    rt[ROWS * NPART];   // 14336 B
  __shared__ unsigned mbits[ROWS];        //   512 B
  const int tid = threadIdx.x;
  const long long rowBase = (long long)blockIdx.x * ROWS;
  long long rem = (long long)B - rowBase;
  const int rows = rem < ROWS ? (int)rem : ROWS;
  const float thr = *thr_p;

  if (rows == ROWS) {
    const unsigned long long srcX =
        (unsigned long long)(const void*)x + (unsigned long long)rowBase * (FEATS * 4ull);
    const unsigned long long srcR =
        (unsigned long long)(const void*)ru + (unsigned long long)rowBase * (NPART * 4ull);
    const unsigned dstX = lds_off(xt);
    const unsigned dstR = lds_off(rt);
    const int chunksX = ROWS * FEATS * 4 / 16;   // 2720
    const int chunksR = ROWS * NPART * 4 / 16;   // 896
    for (int c = tid; c < chunksX; c += ROWS)
      async_copy_b128(dstX + (unsigned)c * 16u, srcX + (unsigned long long)c * 16u);
    for (int c = tid; c < chunksR; c += ROWS)
      async_copy_b128(dstR + (unsigned)c * 16u, srcR + (unsigned long long)c * 16u);
    __builtin_amdgcn_s_wait_asynccnt(0);
  } else {
    for (int i = tid; i < rows * FEATS; i += ROWS)
      xt[i] = x[rowBase * FEATS + i];
    for (int i = tid; i < rows * NPART; i += ROWS)
      rt[i] = ru[rowBase * NPART + i];
  }
  __syncthreads();

  if (tid < rows) {
    unsigned act = 0; int cnt = 0;
#pragma unroll
    for (int k = 0; k < NPART; ++k) {
      bool a = xt[tid * FEATS + 1 + 3 * k] > 0.01f;
      act |= (unsigned)a << k;
      cnt += a ? 1 : 0;
    }
    const float prob = ((float)cnt > thr) ? 0.3f : 0.05f;
    unsigned m = 0;
#pragma unroll
    for (int k = 0; k < NPART; ++k) {
      bool mk = (((act >> k) & 1u) != 0u) && (rt[tid * NPART + k] < prob);
      m |= (unsigned)mk << k;
    }
    mbits[tid] = m;
  }
  __syncthreads();

  const int n = rows * FEATS;
  for (int i = tid; i < n; i += ROWS) {
    const int r = i / FEATS;
    const int f = i - r * FEATS;
    float v = xt[i];
    if (f != 0) {
      const int p = (f - 1) / 3;             // feature -> particle (closed form)
      if ((mbits[r] >> p) & 1u) v = 0.0f;
    }
    __builtin_nontemporal_store(v, &out[rowBase * FEATS + i]);
  }
}

extern "C" void kernel_launch(void* const* d_in, const int* in_sizes, int n_in,
                              void* d_out, int out_size, void* d_ws, size_t ws_size,
                              hipStream_t stream) {
  const float* x  = (const float*)d_in[0];   // (B, 85) float32
  const float* ru = (const float*)d_in[1];   // (B, 28) float32
  float* out = (float*)d_out;                // (B, 85) float32
  const int B = in_sizes[0] / FEATS;

  unsigned* hist = (unsigned*)d_ws;                     // 29 bins
  float*    thr  = (float*)((char*)d_ws + 128);         // interpolated quantile

  const int blocks = (B + ROWS - 1) / ROWS;
  mm_init  <<<1, 32, 0, stream>>>(hist);
  mm_hist  <<<blocks, ROWS, 0, stream>>>(x, hist, B);
  mm_thresh<<<1, 1, 0, stream>>>(hist, thr, B);
  mm_apply <<<blocks, ROWS, 0, stream>>>(x, ru, thr, out, B);
}